// DSACPatchLayer_76287209111874
// MI455X (gfx1250) — compile-verified
//
#include <hip/hip_runtime.h>
#include <math.h>

#define PATCH 128
#define NHYP  256
#define BATCH 8
#define NPIX  (PATCH*PATCH)   // 16384
#define HPG   16              // hypotheses per group (per block in kernel B)

typedef __attribute__((ext_vector_type(2))) float v2f;
typedef __attribute__((ext_vector_type(8))) float v8f;
typedef __attribute__((ext_vector_type(4))) unsigned int v4u;
typedef __attribute__((ext_vector_type(4))) int v4i;
typedef __attribute__((ext_vector_type(8))) int v8i;

#if defined(__HIP_DEVICE_COMPILE__) && __has_builtin(__builtin_amdgcn_wmma_f32_16x16x4_f32)
#define USE_WMMA_F32X4 1
#else
#define USE_WMMA_F32X4 0
#endif

#if defined(__HIP_DEVICE_COMPILE__) && \
    __has_builtin(__builtin_amdgcn_tensor_load_to_lds) && \
    __has_builtin(__builtin_amdgcn_s_wait_tensorcnt)
#define USE_TDM 1
#else
#define USE_TDM 0
#endif

// ---------------------------------------------------------------------------
// Kernel A: per-hypothesis homography. A is 8x9 with an EXACT null space
// (8 rows), and the reference divides by H[2,2]; so solving A[:, :8] x = -A[:,8]
// with h8 = 1 reproduces the SVD null vector after the reference's
// normalization. Gaussian elimination with partial pivoting, f32.
// ---------------------------------------------------------------------------
__global__ __launch_bounds__(256) void dsac_solve_h(
    const float* __restrict__ noise,   // [B, NHYP, 4, 2]
    const int*   __restrict__ perm,    // [B, NHYP, 4]
    float*       __restrict__ Hout)    // [B*NHYP, 9]
{
  int hyp = blockIdx.x * blockDim.x + threadIdx.x;
  if (hyp >= BATCH * NHYP) return;

  const float bx[5] = {0.f, 128.f, 128.f,   0.f, 64.f};
  const float by[5] = {0.f,   0.f, 128.f, 128.f, 64.f};

  float Aa[8][9];   // cols 0..7 = A[:, :8], col 8 = rhs (= +u / +v)
  const int*   pi = perm  + (size_t)hyp * 4;
  const float* nz = noise + (size_t)hyp * 8;
  for (int j = 0; j < 4; ++j) {
    int p = pi[j];
    float x = bx[p], y = by[p];
    float u = x + 8.0f * nz[2*j+0];
    float v = y + 8.0f * nz[2*j+1];
    float* r1 = Aa[2*j];
    float* r2 = Aa[2*j+1];
    r1[0]=x;   r1[1]=y;   r1[2]=1.f; r1[3]=0.f; r1[4]=0.f; r1[5]=0.f;
    r1[6]=-u*x; r1[7]=-u*y; r1[8]=u;
    r2[0]=0.f; r2[1]=0.f; r2[2]=0.f; r2[3]=x;   r2[4]=y;   r2[5]=1.f;
    r2[6]=-v*x; r2[7]=-v*y; r2[8]=v;
  }

  for (int k = 0; k < 8; ++k) {
    int piv = k; float best = fabsf(Aa[k][k]);
    for (int r = k + 1; r < 8; ++r) {
      float m = fabsf(Aa[r][k]);
      if (m > best) { best = m; piv = r; }
    }
    if (piv != k) {
      for (int c = k; c < 9; ++c) {
        float t = Aa[k][c]; Aa[k][c] = Aa[piv][c]; Aa[piv][c] = t;
      }
    }
    float inv = 1.0f / Aa[k][k];
    for (int r = k + 1; r < 8; ++r) {
      float f = Aa[r][k] * inv;
      for (int c = k; c < 9; ++c) Aa[r][c] = fmaf(-f, Aa[k][c], Aa[r][c]);
    }
  }
  float h[9];
  for (int k = 7; k >= 0; --k) {
    float s = Aa[k][8];
    for (int c = k + 1; c < 8; ++c) s -= Aa[k][c] * h[c];
    h[k] = s / Aa[k][k];
  }
  h[8] = 1.0f;
  // replicate reference's double normalization by (H[2,2] + 1e-8)
  float d1 = h[8] + 1e-8f;
  for (int i = 0; i < 9; ++i) h[i] /= d1;
  float d2 = h[8] + 1e-8f;
  for (int i = 0; i < 9; ++i) h[i] /= d2;

  float* o = Hout + (size_t)hyp * 9;
  for (int i = 0; i < 9; ++i) o[i] = h[i];
}

// ---------------------------------------------------------------------------
// TDM helper: 2D tile load, tile == tensor == 128x128 f32, into LDS.
// D# bitfields per CDNA5 ISA ch.8 (group0: count/lds_addr/global_addr/type,
// group1: data_size/dims/strides). Groups 2-3 zero (<=2D tensor).
// ---------------------------------------------------------------------------
#if USE_TDM
__device__ __forceinline__ void tdm_load_patch(const float* gsrc, unsigned ldsAddr) {
  unsigned long long ga = (unsigned long long)(uintptr_t)gsrc;
  v4u d0;
  d0.x = 1u;                                              // count=1, user desc
  d0.y = ldsAddr;                                         // lds_addr (bytes)
  d0.z = (unsigned)(ga & 0xFFFFFFFFull);                  // global_addr[31:0]
  d0.w = (unsigned)((ga >> 32) & 0x1FFFFFFull) | (2u << 30); // addr[56:32], type=2
  v8i d1;
  d1[0] = (int)(2u << 16);      // data_size=2 (4 bytes), no mask/pad/iterate
  d1[1] = (int)(128u << 16);    // tensor_dim0 = 128 (atomic_barrier_addr = 0)
  d1[2] = (int)(128u << 16);    // tensor_dim1 = 128
  d1[3] = (int)(128u << 16);    // tile_dim0 = 128
  d1[4] = (int)128u;            // tile_dim1 = 128, tile_dim2 = 0
  d1[5] = (int)128u;            // tensor_dim0_stride = 128
  d1[6] = 0;
  d1[7] = 0;
  v4i dz = {0, 0, 0, 0};
#if __clang_major__ >= 23
  v8i dz8 = {0, 0, 0, 0, 0, 0, 0, 0};
  __builtin_amdgcn_tensor_load_to_lds(d0, d1, dz, dz, dz8, 0);
#else
  __builtin_amdgcn_tensor_load_to_lds(d0, d1, dz, dz, 0);
#endif
}
#endif

// ---------------------------------------------------------------------------
// Kernel B: warp + score. Block = (b, group of 16 hyps), 256 threads = 8 waves.
// Both patches live in LDS (128 KB), staged by the Tensor Data Mover.
// Lane layout matches the V_WMMA_F32_16X16X4_F32 A-matrix striping:
// lanes 0-15 = hyp m, K=0,1; lanes 16-31 = hyp m, K=2,3. B = ones, so
// D[m][*] accumulates sum |diff| for 16 hypotheses x 4 pixels per WMMA.
// Two independent accumulators break the WMMA RAW chain.
// ---------------------------------------------------------------------------
__device__ __forceinline__ float sample_diff(
    int p, float t0, float t1, float t2, float t3, float t4, float t5,
    const float* sp1, const float* sp2)
{
  const float kInv = 2.0f / 127.0f;
  float xn = (float)(p & (PATCH - 1)) * kInv - 1.0f;
  float yn = (float)(p >> 7)          * kInv - 1.0f;
  float r0 = fmaf(t0, xn, fmaf(t1, yn, t2));
  float r1 = fmaf(t3, xn, fmaf(t4, yn, t5));
  float gx = fmaf(r0, 63.5f, 63.5f);
  float gy = fmaf(r1, 63.5f, 63.5f);
  float x0 = floorf(gx), y0 = floorf(gy);
  float wx = gx - x0,   wy = gy - y0;
  float sum = 0.0f;
#pragma unroll
  for (int ty = 0; ty < 2; ++ty) {
#pragma unroll
    for (int tx = 0; tx < 2; ++tx) {
      float xi = x0 + (float)tx;
      float yi = y0 + (float)ty;
      bool valid = (xi >= 0.0f) && (xi < (float)PATCH) &&
                   (yi >= 0.0f) && (yi < (float)PATCH);
      int xc = (int)fminf(fmaxf(xi, 0.0f), (float)(PATCH - 1));
      int yc = (int)fminf(fmaxf(yi, 0.0f), (float)(PATCH - 1));
      float w = (tx ? wx : 1.0f - wx) * (ty ? wy : 1.0f - wy);
      float val = sp1[yc * PATCH + xc];
      sum += val * (valid ? w : 0.0f);
    }
  }
  return fabsf(sum - sp2[p]);
}

__global__ __launch_bounds__(256) void dsac_warp_score(
    const float* __restrict__ patches,  // [B, 2, 128, 128]
    const float* __restrict__ Htab,     // [B*NHYP, 9]
    float*       __restrict__ scores)   // [B*NHYP] raw sums of |diff|
{
  extern __shared__ float smem[];
  float* sp1 = smem;            // 16384
  float* sp2 = smem + NPIX;     // 16384
  float* ssc = smem + 2*NPIX;   // 16

  const int b   = blockIdx.x;
  const int g   = blockIdx.y;
  const int tid = threadIdx.x;

  const float* p1g = patches + (size_t)b * 2 * NPIX;
  const float* p2g = p1g + NPIX;

#if USE_TDM
  if (tid < 32) {   // one wave issues both DMA descriptors, then waits
    tdm_load_patch(p1g, (unsigned)(uintptr_t)sp1);
    tdm_load_patch(p2g, (unsigned)(uintptr_t)sp2);
    __builtin_amdgcn_s_wait_tensorcnt((short)0);
  }
#else
  __builtin_prefetch(p1g, 0, 1);
  {
    const float4* s1g = (const float4*)p1g;
    const float4* s2g = (const float4*)p2g;
    float4* s1v = (float4*)sp1;
    float4* s2v = (float4*)sp2;
    for (int i = tid; i < NPIX / 4; i += 256) { s1v[i] = s1g[i]; s2v[i] = s2g[i]; }
  }
#endif
  if (tid < HPG) ssc[tid] = 0.0f;
  __syncthreads();

  const int lane = tid & 31;
  const int wav  = tid >> 5;
  const int hloc = lane & 15;   // hypothesis within group (row m of WMMA A)
  const int half = lane >> 4;   // 0 -> K=0,1 ; 1 -> K=2,3

  const float* Hr = Htab + ((size_t)b * NHYP + g * HPG + hloc) * 9;
  float t0 = Hr[0], t1 = Hr[1], t2 = Hr[2];
  float t3 = Hr[3], t4 = Hr[4], t5 = Hr[5];

  const int pixBase = wav * (NPIX / 8);   // 2048 pixels per wave

#if USE_WMMA_F32X4
  v8f acc0 = {0.f,0.f,0.f,0.f,0.f,0.f,0.f,0.f};
  v8f acc1 = {0.f,0.f,0.f,0.f,0.f,0.f,0.f,0.f};
  const v2f ones = {1.0f, 1.0f};
#else
  float facc = 0.0f;
#endif

  for (int it = 0; it < (NPIX / 8) / 8; ++it) {   // 8 pixels/wave/iter
    int pA = pixBase + it * 8 + half * 2;         // pixels 0..3 of octet
    int pB = pA + 4;                              // pixels 4..7 of octet
    float dA0 = sample_diff(pA,     t0,t1,t2,t3,t4,t5, sp1, sp2);
    float dA1 = sample_diff(pA + 1, t0,t1,t2,t3,t4,t5, sp1, sp2);
    float dB0 = sample_diff(pB,     t0,t1,t2,t3,t4,t5, sp1, sp2);
    float dB1 = sample_diff(pB + 1, t0,t1,t2,t3,t4,t5, sp1, sp2);
#if USE_WMMA_F32X4
    v2f a0; a0.x = dA0; a0.y = dA1;
    v2f a1; a1.x = dB0; a1.y = dB1;
    acc0 = __builtin_amdgcn_wmma_f32_16x16x4_f32(
        false, a0, false, ones, (short)0, acc0, false, false);
    acc1 = __builtin_amdgcn_wmma_f32_16x16x4_f32(
        false, a1, false, ones, (short)0, acc1, false, false);
#else
    facc += (dA0 + dA1) + (dB0 + dB1);
#endif
  }

#if USE_WMMA_F32X4
  v8f acc = acc0 + acc1;
  // C/D layout: lane = column N; VGPR j holds row j (lanes 0-15) or row j+8
  // (lanes 16-31). All columns identical (B = ones) -> read column 0.
  if (lane == 0) {
#pragma unroll
    for (int j = 0; j < 8; ++j) atomicAdd(&ssc[j], acc[j]);
  } else if (lane == 16) {
#pragma unroll
    for (int j = 0; j < 8; ++j) atomicAdd(&ssc[8 + j], acc[j]);
  }
#else
  atomicAdd(&ssc[hloc], facc);
#endif
  __syncthreads();
  if (tid < HPG) scores[(size_t)b * NHYP + g * HPG + tid] = ssc[tid];
}

// ---------------------------------------------------------------------------
// Kernel C: softmax over 256 hypotheses, probs out, first-index argmax,
// copy selected H. One block per batch.
// ---------------------------------------------------------------------------
__global__ __launch_bounds__(256) void dsac_softmax_select(
    const float* __restrict__ Htab,    // [B*NHYP, 9]
    const float* __restrict__ scores,  // [B*NHYP] raw sums
    float*       __restrict__ out)     // [72 selected_H | 2048 probs]
{
  __shared__ float sv[256];
  __shared__ int   si[256];
  const int b = blockIdx.x;
  const int n = threadIdx.x;

  float score = -scores[(size_t)b * NHYP + n] * (1.0f / (float)NPIX);

  // max
  sv[n] = score; __syncthreads();
  for (int s = 128; s > 0; s >>= 1) {
    if (n < s) sv[n] = fmaxf(sv[n], sv[n + s]);
    __syncthreads();
  }
  float mx = sv[0]; __syncthreads();

  // sum of exp
  float e = expf(score - mx);
  sv[n] = e; __syncthreads();
  for (int s = 128; s > 0; s >>= 1) {
    if (n < s) sv[n] += sv[n + s];
    __syncthreads();
  }
  float denom = sv[0]; __syncthreads();

  out[BATCH * 9 + (size_t)b * NHYP + n] = e / denom;

  // argmax (ties -> lowest index, matching jnp.argmax)
  sv[n] = score; si[n] = n; __syncthreads();
  for (int s = 128; s > 0; s >>= 1) {
    if (n < s) {
      float v2 = sv[n + s]; int i2 = si[n + s];
      if (v2 > sv[n] || (v2 == sv[n] && i2 < si[n])) { sv[n] = v2; si[n] = i2; }
    }
    __syncthreads();
  }
  if (n == 0) {
    int idx = si[0];
    const float* Hs = Htab + ((size_t)b * NHYP + idx) * 9;
    float* Ho = out + (size_t)b * 9;
    for (int i = 0; i < 9; ++i) Ho[i] = Hs[i];
  }
}

// ---------------------------------------------------------------------------
extern "C" void kernel_launch(void* const* d_in, const int* in_sizes, int n_in,
                              void* d_out, int out_size, void* d_ws, size_t ws_size,
                              hipStream_t stream) {
  const float* patches = (const float*)d_in[0];  // [8,2,128,128] f32
  const float* noise   = (const float*)d_in[1];  // [8,256,4,2]   f32
  const int*   perm    = (const int*)  d_in[2];  // [8,256,4]     i32
  float* out = (float*)d_out;

  float* wsH = (float*)d_ws;                         // 2048 * 9 floats
  float* wsS = wsH + (size_t)BATCH * NHYP * 9;       // 2048 floats

  dsac_solve_h<<<dim3((BATCH * NHYP) / 256), dim3(256), 0, stream>>>(noise, perm, wsH);

  size_t shmem = (size_t)(2 * NPIX + HPG) * sizeof(float);  // 128 KB + 64 B
  dsac_warp_score<<<dim3(BATCH, NHYP / HPG), dim3(256), shmem, stream>>>(patches, wsH, wsS);

  dsac_softmax_select<<<dim3(BATCH), dim3(256), 0, stream>>>(wsH, wsS, out);
}